// CenterDirectionLoss_15272903704671
// MI455X (gfx1250) — compile-verified
//
#include <hip/hip_runtime.h>

#define HDIM 768
#define WDIM 768
#define HW (HDIM * WDIM)          // 589824
#define HW4 (HW / 4)              // 147456 vec4 elements
#define BATCH 8
#define MAXI 33
#define GRIDX 144                 // 147456 / (144*256) = 4 iterations, exact
#define BLOCK 256

typedef __attribute__((ext_vector_type(2))) float v2f;
typedef __attribute__((ext_vector_type(8))) float v8f;

// Exact f32 wave(32)-sum using V_WMMA_F32_16X16X4_F32.
// A[m][0] = x(lane m), A[m][2] = x(lane m+16), rest 0; B = ones; C = 0.
// => D[m][n] = x_m + x_{m+16}. Lane n holds D[0..7][n] (lanes 16..31 hold
// D[8..15][n]) in 8 VGPRs; sum those and shfl_xor(16) for the full wave sum.
__device__ __forceinline__ float wave_sum_wmma(float x) {
    v2f a; a[0] = x;    a[1] = 0.0f;
    v2f o; o[0] = 1.0f; o[1] = 1.0f;
    v8f c = {};
    v8f d = __builtin_amdgcn_wmma_f32_16x16x4_f32(
        false, a, false, o, (short)0, c, false, false);
    float s = d[0] + d[1] + d[2] + d[3] + d[4] + d[5] + d[6] + d[7];
    s += __shfl_xor(s, 16, 32);
    return s;
}

__global__ void cdl_init_kernel(unsigned int* __restrict__ counts) {
    for (int i = threadIdx.x; i < BATCH * MAXI; i += BLOCK) counts[i] = 0u;
}

__global__ void cdl_hist_kernel(const int* __restrict__ inst,
                                const int* __restrict__ ign,
                                unsigned int* __restrict__ gcounts) {
    __shared__ unsigned int lc[MAXI];
    const int b = blockIdx.y;
    for (int i = threadIdx.x; i < MAXI; i += BLOCK) lc[i] = 0u;
    __syncthreads();
    const int4* inst4 = (const int4*)(inst + b * HW);
    const int4* ign4  = (const int4*)(ign  + b * HW);
    for (int q = blockIdx.x * BLOCK + threadIdx.x; q < HW4; q += GRIDX * BLOCK) {
        int4 id = inst4[q];
        int4 ig = ign4[q];
        if (ig.x == 0) atomicAdd(&lc[id.x], 1u);
        if (ig.y == 0) atomicAdd(&lc[id.y], 1u);
        if (ig.z == 0) atomicAdd(&lc[id.z], 1u);
        if (ig.w == 0) atomicAdd(&lc[id.w], 1u);
    }
    __syncthreads();
    for (int i = threadIdx.x; i < MAXI; i += BLOCK) {
        unsigned int v = lc[i];
        if (v) atomicAdd(&gcounts[b * MAXI + i], v);
    }
}

__global__ void cdl_loss_kernel(const float* __restrict__ pred,
                                const int* __restrict__ inst,
                                const float* __restrict__ gt_R,
                                const float* __restrict__ gt_sin,
                                const float* __restrict__ gt_cos,
                                const float* __restrict__ gci,
                                const int* __restrict__ ign,
                                const unsigned int* __restrict__ gcounts,
                                float* __restrict__ partials) {
    __shared__ float s_cnt[MAXI];
    __shared__ float s_w[MAXI];
    __shared__ float s_ninst;
    __shared__ float s_red[BLOCK / 32][3];

    const int b = blockIdx.y;
    const int tid = threadIdx.x;

    if (tid < MAXI) s_cnt[tid] = (float)gcounts[b * MAXI + tid];
    __syncthreads();
    if (tid == 0) {
        float n = 0.0f;
        for (int k = 1; k < MAXI; k++)
            if (s_cnt[k] > 0.0f) n += 1.0f;
        s_ninst = fmaxf(n, 1.0f);
    }
    __syncthreads();
    if (tid < MAXI) {
        float sz = fmaxf(s_cnt[tid], 1.0f);
        s_w[tid] = (tid == 0) ? (1.0f / sz) : (1.0f / (sz * s_ninst));
    }
    __syncthreads();

    const size_t boff = (size_t)b * HW;
    const int4*   inst4 = (const int4*)  (inst   + boff);
    const int4*   ign4  = (const int4*)  (ign    + boff);
    const float4* gci4  = (const float4*)(gci    + boff);
    const float4* gs4   = (const float4*)(gt_sin + boff);
    const float4* gc4   = (const float4*)(gt_cos + boff);
    const float4* gr4   = (const float4*)(gt_R   + boff);
    const float4* ps4   = (const float4*)(pred + (size_t)b * 3 * HW);
    const float4* pc4   = ps4 + HW4;
    const float4* pr4   = ps4 + 2 * HW4;

    float as = 0.0f, ac = 0.0f, ar = 0.0f;
    // Uniform trip count (4 iters for every thread): EXEC stays all-1s.
    for (int q = blockIdx.x * BLOCK + tid; q < HW4; q += GRIDX * BLOCK) {
        int4   id = inst4[q];
        int4   ig = ign4[q];
        float4 gc = gci4[q];
        float4 gs = gs4[q];
        float4 gg = gc4[q];
        float4 gr = gr4[q];
        float4 s  = ps4[q];
        float4 c  = pc4[q];
        float4 r  = pr4[q];

        // Branchless: unconditional LDS lookup * {0,1} valid * center-ignore.
        float m0 = s_w[id.x] * ((ig.x == 0) ? 1.0f : 0.0f) * gc.x;
        float m1 = s_w[id.y] * ((ig.y == 0) ? 1.0f : 0.0f) * gc.y;
        float m2 = s_w[id.z] * ((ig.z == 0) ? 1.0f : 0.0f) * gc.z;
        float m3 = s_w[id.w] * ((ig.w == 0) ? 1.0f : 0.0f) * gc.w;

        as += m0 * fabsf(s.x - gs.x) + m1 * fabsf(s.y - gs.y)
            + m2 * fabsf(s.z - gs.z) + m3 * fabsf(s.w - gs.w);
        ac += m0 * fabsf(c.x - gg.x) + m1 * fabsf(c.y - gg.y)
            + m2 * fabsf(c.z - gg.z) + m3 * fabsf(c.w - gg.w);
        ar += m0 * fabsf(r.x - log1pf(gr.x)) + m1 * fabsf(r.y - log1pf(gr.y))
            + m2 * fabsf(r.z - log1pf(gr.z)) + m3 * fabsf(r.w - log1pf(gr.w));
    }

    float ws = wave_sum_wmma(as);
    float wc = wave_sum_wmma(ac);
    float wr = wave_sum_wmma(ar);

    const int wave = tid >> 5;
    if ((tid & 31) == 0) {
        s_red[wave][0] = ws; s_red[wave][1] = wc; s_red[wave][2] = wr;
    }
    __syncthreads();
    if (tid == 0) {
        float t0 = 0.0f, t1 = 0.0f, t2 = 0.0f;
        for (int w2 = 0; w2 < BLOCK / 32; w2++) {
            t0 += s_red[w2][0]; t1 += s_red[w2][1]; t2 += s_red[w2][2];
        }
        partials[(b * 3 + 0) * GRIDX + blockIdx.x] = t0;
        partials[(b * 3 + 1) * GRIDX + blockIdx.x] = t1;
        partials[(b * 3 + 2) * GRIDX + blockIdx.x] = t2;
    }
}

__global__ void cdl_final_kernel(const float* __restrict__ partials,
                                 float* __restrict__ out) {
    __shared__ float s[BATCH][3];
    const int tid = threadIdx.x;
    if (tid < BATCH * 3) {
        int b = tid / 3, j = tid % 3;
        const float* p = partials + (b * 3 + j) * GRIDX;
        float acc = 0.0f;
        for (int k = 0; k < GRIDX; k++) acc += p[k];  // fixed order: deterministic
        s[b][j] = acc;
    }
    __syncthreads();
    if (tid < BATCH) {
        float ls = s[tid][0], lc = s[tid][1], lr = s[tid][2];
        float tot = ls + lc + lr;
        out[0 * BATCH + tid] = tot;   // loss
        out[1 * BATCH + tid] = 0.0f;  // zeros
        out[2 * BATCH + tid] = tot;   // loss_centerdir_total
        out[3 * BATCH + tid] = 0.0f;  // zeros
        out[4 * BATCH + tid] = ls;    // loss_sin
        out[5 * BATCH + tid] = lc;    // loss_cos
        out[6 * BATCH + tid] = lr;    // loss_r
        out[7 * BATCH + tid] = 0.0f;  // zeros
    }
}

extern "C" void kernel_launch(void* const* d_in, const int* in_sizes, int n_in,
                              void* d_out, int out_size, void* d_ws, size_t ws_size,
                              hipStream_t stream) {
    const float* pred   = (const float*)d_in[0];  // [8,3,768,768]
    const int*   inst   = (const int*)  d_in[1];  // [8,1,768,768]
    // d_in[2] = label (unused by the reference loss)
    const float* gt_R   = (const float*)d_in[3];
    const float* gt_sin = (const float*)d_in[4];
    const float* gt_cos = (const float*)d_in[5];
    const float* gci    = (const float*)d_in[6];
    const int*   ign    = (const int*)  d_in[7];

    unsigned int* counts   = (unsigned int*)d_ws;           // 8*33 u32
    float*        partials = (float*)((char*)d_ws + 4096);  // 8*3*144 f32

    cdl_init_kernel<<<1, BLOCK, 0, stream>>>(counts);
    dim3 grid(GRIDX, BATCH);
    cdl_hist_kernel<<<grid, BLOCK, 0, stream>>>(inst, ign, counts);
    cdl_loss_kernel<<<grid, BLOCK, 0, stream>>>(pred, inst, gt_R, gt_sin, gt_cos,
                                                gci, ign, counts, partials);
    cdl_final_kernel<<<1, 64, 0, stream>>>(partials, (float*)d_out);
}